// LSTMFinder_13915694039499
// MI455X (gfx1250) — compile-verified
//
#include <hip/hip_runtime.h>
#include <math.h>

#define E 128
#define W 64
#define C 1024
#define S 8

typedef float v2f __attribute__((ext_vector_type(2)));
typedef float v8f __attribute__((ext_vector_type(8)));

// ---------------------------------------------------------------- init
__global__ void init_state(float* __restrict__ h, float* __restrict__ c,
                           int* __restrict__ cur, const int* __restrict__ start_ent) {
    int i = blockIdx.x * blockDim.x + threadIdx.x;
    if (i < W * E) { h[i] = 0.f; c[i] = 0.f; }
    if (i < W) cur[i] = start_ent[i];
}

// ------------------------------------------- feat = [ent_emb[cur] | h | relation]
__global__ void gather_feat(const float* __restrict__ ent_emb,
                            const float* __restrict__ relation,
                            const int* __restrict__ cur,
                            const float* __restrict__ h,
                            float* __restrict__ feat) {
    int i = blockIdx.x * blockDim.x + threadIdx.x;   // W * 3E
    if (i >= W * 3 * E) return;
    int w = i / (3 * E), j = i % (3 * E);
    float v;
    if (j < E)          v = ent_emb[(size_t)cur[w] * E + j];
    else if (j < 2 * E) v = h[w * E + (j - E)];
    else                v = relation[j - 2 * E];
    feat[i] = v;
}

// ---------------------------------------------------------------- WMMA GEMM
// D[M,N] = act( A1[M,K1] @ B1[K1,N] (+ A2[M,K2] @ B2[K2,N]) + bias[N] )
// Row-major. M,N multiples of 16; K multiples of 4. One wave per 16x16 tile.
// Uses V_WMMA_F32_16X16X4_F32 (fp32 in/out — matches reference numerics).
__global__ void wmma_gemm(const float* __restrict__ A1, const float* __restrict__ B1, int K1,
                          const float* __restrict__ A2, const float* __restrict__ B2, int K2,
                          const float* __restrict__ bias, float* __restrict__ D,
                          int M, int N, int relu) {
    const int lane = threadIdx.x & 31;
    const int wave = (blockIdx.x * blockDim.x + threadIdx.x) >> 5;
    const int ntn = N >> 4;
    const int ntm = M >> 4;
    if (wave >= ntm * ntn) return;               // wave-uniform
    const int tm = wave / ntn, tn = wave % ntn;

    const int r    = lane & 15;                  // A row-in-tile == B/D col-in-tile
    const int koff = (lane >> 4) << 1;           // lanes 0-15 -> K {0,1}; 16-31 -> {2,3}

    v8f acc;
    float bv = bias ? bias[tn * 16 + r] : 0.f;   // bias depends on column only
#pragma unroll
    for (int q = 0; q < 8; ++q) acc[q] = bv;

    const float* arow = A1 + (size_t)(tm * 16 + r) * K1;
    for (int kb = 0; kb < K1; kb += 4) {
        v2f a = *(const v2f*)(arow + kb + koff);
        v2f b;
        b[0] = B1[(size_t)(kb + koff)     * N + tn * 16 + r];
        b[1] = B1[(size_t)(kb + koff + 1) * N + tn * 16 + r];
        acc = __builtin_amdgcn_wmma_f32_16x16x4_f32(false, a, false, b,
                                                    (short)0, acc, false, false);
    }
    if (A2) {                                    // fused second matmul (LSTM recurrent)
        const float* arow2 = A2 + (size_t)(tm * 16 + r) * K2;
        for (int kb = 0; kb < K2; kb += 4) {
            v2f a = *(const v2f*)(arow2 + kb + koff);
            v2f b;
            b[0] = B2[(size_t)(kb + koff)     * N + tn * 16 + r];
            b[1] = B2[(size_t)(kb + koff + 1) * N + tn * 16 + r];
            acc = __builtin_amdgcn_wmma_f32_16x16x4_f32(false, a, false, b,
                                                        (short)0, acc, false, false);
        }
    }
    const int rbase = tm * 16 + ((lane >> 4) << 3);
    const int col   = tn * 16 + r;
#pragma unroll
    for (int q = 0; q < 8; ++q) {
        float v = acc[q];
        if (relu) v = v > 0.f ? v : 0.f;
        D[(size_t)(rbase + q) * N + col] = v;
    }
}

// ----------------------------------- per-beam candidate scores + softmax -> probs
// block = one beam (256 threads, 4 candidates each); gather-bound.
// The per-beam feature vector is staged into LDS with the CDNA5 async
// memory->LDS path (GLOBAL_LOAD_ASYNC_TO_LDS_B32, drained via s_wait_asynccnt).
__global__ void scores_softmax(const float* __restrict__ rel_emb,
                               const float* __restrict__ ent_emb,
                               const int* __restrict__ cr,
                               const int* __restrict__ ce,
                               const float* __restrict__ feature,
                               float* __restrict__ probs) {
    __shared__ float sf[2 * E];
    __shared__ float sc[C];
    __shared__ float red[256];
    const int w = blockIdx.x;
    const int t = threadIdx.x;

    {
        // LDS aperture maps addr[31:0] -> LDS offset; static __shared__ starts at as(3) 0.
        unsigned lds_off = (unsigned)(size_t)&sf[t];
        const float* gsrc = feature + w * 2 * E + t;
        asm volatile("global_load_async_to_lds_b32 %0, %1, off"
                     :: "v"(lds_off), "v"(gsrc) : "memory");
        asm volatile("s_wait_asynccnt 0x0" ::: "memory");
    }
    __syncthreads();

    float loc[4];
#pragma unroll
    for (int u = 0; u < 4; ++u) {
        int cdx = t + u * 256;
        int rid = cr[w * C + cdx];
        int eid = ce[w * C + cdx];
        if (u < 3) {   // prefetch next candidate's rows while we compute this dot
            __builtin_prefetch(rel_emb + (size_t)cr[w * C + cdx + 256] * E, 0, 0);
            __builtin_prefetch(ent_emb + (size_t)ce[w * C + cdx + 256] * E, 0, 0);
        }
        const float4* rr = (const float4*)(rel_emb + (size_t)rid * E);
        const float4* er = (const float4*)(ent_emb + (size_t)eid * E);
        float acc = 0.f;
#pragma unroll 8
        for (int q = 0; q < E / 4; ++q) {
            float4 rv = rr[q];
            acc += rv.x * sf[4*q] + rv.y * sf[4*q+1] + rv.z * sf[4*q+2] + rv.w * sf[4*q+3];
        }
#pragma unroll 8
        for (int q = 0; q < E / 4; ++q) {
            float4 ev = er[q];
            acc += ev.x * sf[E+4*q] + ev.y * sf[E+4*q+1] + ev.z * sf[E+4*q+2] + ev.w * sf[E+4*q+3];
        }
        sc[cdx] = acc;
        loc[u]  = acc;
    }
    // block max over 1024
    float m = fmaxf(fmaxf(loc[0], loc[1]), fmaxf(loc[2], loc[3]));
    red[t] = m;
    __syncthreads();
    for (int off = 128; off > 0; off >>= 1) {
        if (t < off) red[t] = fmaxf(red[t], red[t + off]);
        __syncthreads();
    }
    m = red[0];
    __syncthreads();
    // exp + sum
    float lsum = 0.f;
#pragma unroll
    for (int u = 0; u < 4; ++u) {
        int cdx = t + u * 256;
        float e = expf(sc[cdx] - m);
        sc[cdx] = e;
        lsum += e;
    }
    red[t] = lsum;
    __syncthreads();
    for (int off = 128; off > 0; off >>= 1) {
        if (t < off) red[t] += red[t + off];
        __syncthreads();
    }
    float inv = 1.f / red[0];
#pragma unroll
    for (int u = 0; u < 4; ++u) {
        int cdx = t + u * 256;
        probs[(size_t)w * C + cdx] = sc[cdx] * inv;
    }
}

// ------------------------- global top-64 over W*C probs (single block, 1024 thr)
// Iterative block argmax with LDS selection bitmap; ties -> lowest index
// (matches lax.top_k stable descending order).
__global__ void topk64(const float* __restrict__ probs,
                       const int* __restrict__ ce,
                       int* __restrict__ parent, int* __restrict__ sel,
                       int* __restrict__ cur) {
    __shared__ unsigned int bitmap[(W * C) / 32];   // 8 KB
    __shared__ float rv[1024];
    __shared__ int   ri[1024];
    const int t = threadIdx.x;
    bitmap[t] = 0u; bitmap[t + 1024] = 0u;
    __syncthreads();
    for (int it = 0; it < W; ++it) {
        float best = -INFINITY; int bidx = 0x7fffffff;
        for (int e = 0; e < (W * C) / 1024; ++e) {
            int idx = t + e * 1024;                 // ascending -> first max = lowest idx
            if (!((bitmap[idx >> 5] >> (idx & 31)) & 1u)) {
                float v = probs[idx];
                if (v > best) { best = v; bidx = idx; }
            }
        }
        rv[t] = best; ri[t] = bidx;
        __syncthreads();
        for (int off = 512; off > 0; off >>= 1) {
            if (t < off) {
                float v2 = rv[t + off]; int i2 = ri[t + off];
                if (v2 > rv[t] || (v2 == rv[t] && i2 < ri[t])) { rv[t] = v2; ri[t] = i2; }
            }
            __syncthreads();
        }
        if (t == 0) {
            int idx = ri[0];
            parent[it] = idx / C;
            sel[it]    = idx % C;
            cur[it]    = ce[idx];                   // cur_new = ce[parent, sel]
            bitmap[idx >> 5] |= (1u << (idx & 31));
        }
        __syncthreads();
    }
}

// ---------------- gather selected candidate embedding + parent h/c for LSTM
__global__ void gather_lstm(const float* __restrict__ rel_emb,
                            const float* __restrict__ ent_emb,
                            const int* __restrict__ cr, const int* __restrict__ ce,
                            const int* __restrict__ parent, const int* __restrict__ sel,
                            const float* __restrict__ h, const float* __restrict__ c,
                            float* __restrict__ inp,
                            float* __restrict__ hp, float* __restrict__ cp) {
    int i = blockIdx.x * blockDim.x + threadIdx.x;   // W * 2E
    if (i >= W * 2 * E) return;
    int w = i / (2 * E), j = i % (2 * E);
    int p = parent[w], q = sel[w];
    float v;
    if (j < E) v = rel_emb[(size_t)cr[p * C + q] * E + j];
    else       v = ent_emb[(size_t)ce[p * C + q] * E + (j - E)];
    inp[i] = v;
    if (j < E) {
        hp[w * E + j] = h[p * E + j];
        cp[w * E + j] = c[p * E + j];
    }
}

// ---------------------------------------------------------------- LSTM gates
__global__ void lstm_gates(const float* __restrict__ z, const float* __restrict__ cp,
                           float* __restrict__ h, float* __restrict__ c) {
    int i = blockIdx.x * blockDim.x + threadIdx.x;   // W * E
    if (i >= W * E) return;
    int w = i / E, j = i % E;
    const float* zr = z + (size_t)w * 4 * E;
    float zi = zr[j], zf = zr[E + j], zg = zr[2 * E + j], zo = zr[3 * E + j];
    float si = 1.f / (1.f + expf(-zi));
    float sf = 1.f / (1.f + expf(-zf));
    float so = 1.f / (1.f + expf(-zo));
    float cn = sf * cp[i] + si * tanhf(zg);
    float hn = so * tanhf(cn);
    c[i] = cn;
    h[i] = hn;    // hist == h in the reference carry
}

// ---------------------------------------------------------------- launcher
extern "C" void kernel_launch(void* const* d_in, const int* in_sizes, int n_in,
                              void* d_out, int out_size, void* d_ws, size_t ws_size,
                              hipStream_t stream) {
    const float* ent_emb   = (const float*)d_in[0];
    const float* rel_emb   = (const float*)d_in[1];
    const float* relation  = (const float*)d_in[2];
    const float* W1        = (const float*)d_in[3];
    const float* b1        = (const float*)d_in[4];
    const float* W2        = (const float*)d_in[5];
    const float* b2        = (const float*)d_in[6];
    const float* lstm_k    = (const float*)d_in[7];
    const float* lstm_r    = (const float*)d_in[8];
    const float* lstm_b    = (const float*)d_in[9];
    const int*   cand_rel  = (const int*)d_in[10];
    const int*   cand_ent  = (const int*)d_in[11];
    const int*   start_ent = (const int*)d_in[12];

    float* p = (float*)d_ws;
    float* feat  = p;  p += W * 3 * E;   // 24576
    float* hmid  = p;  p += W * 3 * E;   // 24576
    float* featr = p;  p += W * 2 * E;   // 16384
    float* hbuf  = p;  p += W * E;       // 8192 (h == hist)
    float* cbuf  = p;  p += W * E;
    float* hpbuf = p;  p += W * E;
    float* cpbuf = p;  p += W * E;
    float* inp   = p;  p += W * 2 * E;
    float* zbuf  = p;  p += W * 4 * E;   // 32768
    int* cur    = (int*)p;
    int* parent = cur + W;
    int* sel    = parent + W;

    init_state<<<(W * E + 255) / 256, 256, 0, stream>>>(hbuf, cbuf, cur, start_ent);

    for (int s = 0; s < S; ++s) {
        const int* cr = cand_rel + (size_t)s * W * C;
        const int* ce = cand_ent + (size_t)s * W * C;
        float* probs  = (float*)d_out + (size_t)s * W * C;

        gather_feat<<<(W * 3 * E + 255) / 256, 256, 0, stream>>>(ent_emb, relation, cur, hbuf, feat);
        // hmid = relu(feat @ W1 + b1) : 64x384 @ 384x384  (96 tiles -> 12 blocks of 8 waves)
        wmma_gemm<<<12, 256, 0, stream>>>(feat, W1, 3 * E, nullptr, nullptr, 0,
                                          b1, hmid, W, 3 * E, 1);
        // feature = relu(hmid @ W2 + b2) : 64x384 @ 384x256 (64 tiles -> 8 blocks)
        wmma_gemm<<<8, 256, 0, stream>>>(hmid, W2, 3 * E, nullptr, nullptr, 0,
                                         b2, featr, W, 2 * E, 1);
        scores_softmax<<<W, 256, 0, stream>>>(rel_emb, ent_emb, cr, ce, featr, probs);
        topk64<<<1, 1024, 0, stream>>>(probs, ce, parent, sel, cur);
        gather_lstm<<<(W * 2 * E + 255) / 256, 256, 0, stream>>>(rel_emb, ent_emb, cr, ce,
                                                                 parent, sel, hbuf, cbuf,
                                                                 inp, hpbuf, cpbuf);
        // z = inp @ lstm_k + hp @ lstm_r + lstm_b : 64x512 out (128 tiles -> 16 blocks)
        wmma_gemm<<<16, 256, 0, stream>>>(inp, lstm_k, 2 * E, hpbuf, lstm_r, E,
                                          lstm_b, zbuf, W, 4 * E, 0);
        lstm_gates<<<(W * E + 255) / 256, 256, 0, stream>>>(zbuf, cpbuf, hbuf, cbuf);
    }
}